// CrossAttention_35407710388638
// MI455X (gfx1250) — compile-verified
//
#include <hip/hip_runtime.h>
#include <math.h>

// Problem constants (from setup_inputs)
#define BB  2
#define QQ  900
#define CC  256
#define NCAM 6
#define BQ  (BB * QQ)        // 1800
#define BNQ (BB * NCAM * QQ) // 10800

typedef __attribute__((ext_vector_type(2))) float v2f;
typedef __attribute__((ext_vector_type(8))) float v8f;

__device__ __forceinline__ float n2n(float v, float nanv, float pinf, float ninf) {
  if (isnan(v)) return nanv;
  if (isinf(v)) return v > 0.f ? pinf : ninf;
  return v;
}

// Standard adjugate 4x4 inverse (row-major)
__device__ void invert4x4(const float m[16], float inv[16]) {
  inv[0]  =  m[5]*m[10]*m[15] - m[5]*m[11]*m[14] - m[9]*m[6]*m[15] + m[9]*m[7]*m[14] + m[13]*m[6]*m[11] - m[13]*m[7]*m[10];
  inv[4]  = -m[4]*m[10]*m[15] + m[4]*m[11]*m[14] + m[8]*m[6]*m[15] - m[8]*m[7]*m[14] - m[12]*m[6]*m[11] + m[12]*m[7]*m[10];
  inv[8]  =  m[4]*m[9]*m[15]  - m[4]*m[11]*m[13] - m[8]*m[5]*m[15] + m[8]*m[7]*m[13] + m[12]*m[5]*m[11] - m[12]*m[7]*m[9];
  inv[12] = -m[4]*m[9]*m[14]  + m[4]*m[10]*m[13] + m[8]*m[5]*m[14] - m[8]*m[6]*m[13] - m[12]*m[5]*m[10] + m[12]*m[6]*m[9];
  inv[1]  = -m[1]*m[10]*m[15] + m[1]*m[11]*m[14] + m[9]*m[2]*m[15] - m[9]*m[3]*m[14] - m[13]*m[2]*m[11] + m[13]*m[3]*m[10];
  inv[5]  =  m[0]*m[10]*m[15] - m[0]*m[11]*m[14] - m[8]*m[2]*m[15] + m[8]*m[3]*m[14] + m[12]*m[2]*m[11] - m[12]*m[3]*m[10];
  inv[9]  = -m[0]*m[9]*m[15]  + m[0]*m[11]*m[13] + m[8]*m[1]*m[15] - m[8]*m[3]*m[13] - m[12]*m[1]*m[11] + m[12]*m[3]*m[9];
  inv[13] =  m[0]*m[9]*m[14]  - m[0]*m[10]*m[13] - m[8]*m[1]*m[14] + m[8]*m[2]*m[13] + m[12]*m[1]*m[10] - m[12]*m[2]*m[9];
  inv[2]  =  m[1]*m[6]*m[15]  - m[1]*m[7]*m[14]  - m[5]*m[2]*m[15] + m[5]*m[3]*m[14] + m[13]*m[2]*m[7]  - m[13]*m[3]*m[6];
  inv[6]  = -m[0]*m[6]*m[15]  + m[0]*m[7]*m[14]  + m[4]*m[2]*m[15] - m[4]*m[3]*m[14] - m[12]*m[2]*m[7]  + m[12]*m[3]*m[6];
  inv[10] =  m[0]*m[5]*m[15]  - m[0]*m[7]*m[13]  - m[4]*m[1]*m[15] + m[4]*m[3]*m[13] + m[12]*m[1]*m[7]  - m[12]*m[3]*m[5];
  inv[14] = -m[0]*m[5]*m[14]  + m[0]*m[6]*m[13]  + m[4]*m[1]*m[14] - m[4]*m[2]*m[13] - m[12]*m[1]*m[6]  + m[12]*m[2]*m[5];
  inv[3]  = -m[1]*m[6]*m[11]  + m[1]*m[7]*m[10]  + m[5]*m[2]*m[11] - m[5]*m[3]*m[10] - m[9]*m[2]*m[7]   + m[9]*m[3]*m[6];
  inv[7]  =  m[0]*m[6]*m[11]  - m[0]*m[7]*m[10]  - m[4]*m[2]*m[11] + m[4]*m[3]*m[10] + m[8]*m[2]*m[7]   - m[8]*m[3]*m[6];
  inv[11] = -m[0]*m[5]*m[11]  + m[0]*m[7]*m[9]   + m[4]*m[1]*m[11] - m[4]*m[3]*m[9]  - m[8]*m[1]*m[7]   + m[8]*m[3]*m[5];
  inv[15] =  m[0]*m[5]*m[10]  - m[0]*m[6]*m[9]   - m[4]*m[1]*m[10] + m[4]*m[2]*m[9]  + m[8]*m[1]*m[6]   - m[8]*m[2]*m[5];
  float det = m[0]*inv[0] + m[1]*inv[4] + m[2]*inv[8] + m[3]*inv[12];
  float d = 1.0f / det; // det==0 -> inf/nan, cleaned by n2n in caller
  for (int i = 0; i < 16; i++) inv[i] *= d;
}

// ---------------- Kernel 0: project reference points to image plane ----------------
__global__ void proj_kernel(const float* __restrict__ rp, const int* __restrict__ mask,
                            const float* __restrict__ intr, const float* __restrict__ extr,
                            float4* __restrict__ proj) {
  int tid = blockIdx.x * blockDim.x + threadIdx.x;
  if (tid >= BNQ) return;
  int q = tid % QQ;
  int n = (tid / QQ) % NCAM;
  int b = tid / (QQ * NCAM);

  bool msk = mask[b * QQ + q] != 0;
  float r[3];
  #pragma unroll
  for (int k = 0; k < 3; k++) {
    float v = rp[(b * QQ + q) * 3 + k] * 102.4f - 51.2f;
    r[k] = msk ? -1000.f : v;
  }

  float m[16], inv[16];
  const float* e = extr + (size_t)(b * NCAM + n) * 16;
  #pragma unroll
  for (int i = 0; i < 16; i++) m[i] = e[i];
  invert4x4(m, inv);
  #pragma unroll
  for (int i = 0; i < 16; i++) inv[i] = n2n(inv[i], 0.f, 1e6f, -1e6f);

  float pc[4];
  #pragma unroll
  for (int i = 0; i < 4; i++)
    pc[i] = inv[i*4+0]*r[0] + inv[i*4+1]*r[1] + inv[i*4+2]*r[2] + inv[i*4+3];

  float depth = n2n(pc[2], 10.f, 100.f, -100.f);
  bool invalid = depth < 1.5f;
  float ds = fmaxf(depth, 1.5f);
  float p0 = pc[0] / ds, p1 = pc[1] / ds, p2 = pc[2] / ds;

  const float* K = intr + (size_t)(b * NCAM + n) * 9;
  float ix = K[0]*p0 + K[1]*p1 + K[2]*p2;
  float iy = K[3]*p0 + K[4]*p1 + K[5]*p2;
  ix = fminf(fmaxf(ix, -3000.f), 3000.f);
  iy = fminf(fmaxf(iy, -3000.f), 3000.f);
  proj[tid] = make_float4(ix, iy, invalid ? 1.f : 0.f, 0.f);
}

// ---------------- Kernel: bilinear multi-level sampling ----------------
// One block per (b,n,q) point, one lane per channel. Gather-bound.
__global__ void sample_kernel(const float4* __restrict__ proj,
                              const float* __restrict__ f0, const float* __restrict__ f1,
                              const float* __restrict__ f2, const float* __restrict__ f3,
                              float* __restrict__ ms) {
  int bnq = blockIdx.x;
  int c = threadIdx.x;
  float4 p = proj[bnq];
  bool invalid = p.z != 0.f;
  const float* feats[4] = {f0, f1, f2, f3};
  const int Hs[4] = {112, 56, 28, 14};
  const int Wsz[4] = {200, 100, 50, 25};

  float acc = 0.f;
  #pragma unroll
  for (int li = 0; li < 4; li++) {
    const int Hf = Hs[li], Wf = Wsz[li];
    float fx = p.x * ((float)Wf / 800.f);
    float fy = p.y * ((float)Hf / 448.f);
    float gx = fminf(fmaxf(fx / (Wf - 1.f) * 2.f - 1.f, -10.f), 10.f);
    float gy = fminf(fmaxf(fy / (Hf - 1.f) * 2.f - 1.f, -10.f), 10.f);
    if (invalid) { gx = -100.f; gy = -100.f; }
    float px = (gx + 1.f) * 0.5f * (Wf - 1);
    float py = (gy + 1.f) * 0.5f * (Hf - 1);
    float x0f = floorf(px), y0f = floorf(py);
    int x0 = (int)x0f, y0 = (int)y0f;
    float wx1 = px - x0f, wy1 = py - y0f;
    float wx0 = 1.f - wx1, wy0 = 1.f - wy1;

    const float* base = feats[li] + ((size_t)bnq * CC + c) * (size_t)(Hf * Wf);
    float s = 0.f;
    bool vx0 = (x0 >= 0) & (x0 < Wf);
    bool vx1 = (x0 + 1 >= 0) & (x0 + 1 < Wf);
    bool vy0 = (y0 >= 0) & (y0 < Hf);
    bool vy1 = (y0 + 1 >= 0) & (y0 + 1 < Hf);
    if (vy0) {
      const float* row = base + (size_t)y0 * Wf;
      if (vx0) s += row[x0]     * (wx0 * wy0);
      if (vx1) s += row[x0 + 1] * (wx1 * wy0);
    }
    if (vy1) {
      const float* row = base + (size_t)(y0 + 1) * Wf;
      if (vx0) s += row[x0]     * (wx0 * wy1);
      if (vx1) s += row[x0 + 1] * (wx1 * wy1);
    }
    acc += s;
  }
  ms[(size_t)bnq * CC + c] = acc * 0.25f;
}

// ---------------- WMMA f32 GEMM: C[M,256] = A[M,256] @ W^T + bias ----------------
// W is [256,256] row-major, so B[k][n] = W[n][k].
// Tiles are staged into LDS with CDNA5 async global->LDS copies (ASYNCcnt path),
// bypassing VGPRs: global_load_async_to_lds_b128 + s_wait_asynccnt.
// grid: (ceil(M/16), 4); block: 128 (4 waves, each wave one 16x16 N sub-tile).
#define KC  64
#define LDP 68  // padded LDS row stride (272 B = 17*16 B): conflict-free & 16B-aligned

// GVS addressing: 64-bit SGPR base + 32-bit VGPR byte offset; vdst = LDS byte addr.
#define ASYNC_CP_B128(ldsoff, base, goff)                                  \
  asm volatile("global_load_async_to_lds_b128 %0, %1, %2"                  \
               :: "v"(ldsoff), "v"(goff), "s"(base) : "memory")

__global__ void gemm_wmma(const float* __restrict__ A, const float* __restrict__ W,
                          const float* __restrict__ bias, float* __restrict__ Cout,
                          int M, const int* __restrict__ mask) {
  __shared__ float As[16 * LDP];
  __shared__ float Bs[64 * LDP];
  const int K = CC;
  const int t = threadIdx.x;
  const int wave = t >> 5;
  const int lane = t & 31;
  const int baseM = blockIdx.x * 16;
  const int colBase64 = blockIdx.y * 64;
  const int n = lane & 15;           // B/C column within tile
  const int mrow = lane & 15;        // A row within tile
  const int kb = (lane >> 4) * 2;    // K sub-offset per half-wave
  const int col = colBase64 + wave * 16 + n;

  v8f c;
  float bval = bias[col];
  #pragma unroll
  for (int i = 0; i < 8; i++) c[i] = bval;

  for (int k0 = 0; k0 < K; k0 += KC) {
    // Stage A tile (16 x KC) as 256 async b128 copies (2 per thread).
    // Out-of-range M rows are clamped to M-1: they feed only C rows that the
    // guarded store never writes, keeping EXEC uniform for WMMA.
    for (int idx = t; idx < 16 * 16; idx += 128) {
      int r = idx >> 4, c4 = idx & 15;
      int row = baseM + r;
      if (row >= M) row = M - 1;
      unsigned lo = (unsigned)(size_t)(&As[r * LDP + c4 * 4]);
      unsigned go = (unsigned)(((size_t)row * K + k0 + c4 * 4) * sizeof(float));
      ASYNC_CP_B128(lo, A, go);
    }
    // Stage W tile (64 x KC) as 1024 async b128 copies (8 per thread).
    for (int idx = t; idx < 64 * 16; idx += 128) {
      int r = idx >> 4, c4 = idx & 15;
      unsigned lo = (unsigned)(size_t)(&Bs[r * LDP + c4 * 4]);
      unsigned go = (unsigned)(((size_t)(colBase64 + r) * K + k0 + c4 * 4) * sizeof(float));
      ASYNC_CP_B128(lo, W, go);
    }
    asm volatile("s_wait_asynccnt 0" ::: "memory");
    __syncthreads();

    #pragma unroll
    for (int kk = 0; kk < KC; kk += 4) {
      v2f a, b;
      a.x = As[mrow * LDP + kk + kb];
      a.y = As[mrow * LDP + kk + kb + 1];
      b.x = Bs[(wave * 16 + n) * LDP + kk + kb];
      b.y = Bs[(wave * 16 + n) * LDP + kk + kb + 1];
      c = __builtin_amdgcn_wmma_f32_16x16x4_f32(
          /*neg_a=*/false, a, /*neg_b=*/false, b,
          /*c_mod=*/(short)0, c, /*reuse_a=*/false, /*reuse_b=*/false);
    }
    __syncthreads();
  }

  // Store: VGPR r holds rows (r + 8*(lane>>4)), cols = lane&15
  const int rowHalf = (lane >> 4) * 8;
  #pragma unroll
  for (int r = 0; r < 8; r++) {
    int row = baseM + rowHalf + r;
    if (row < M) {
      float val = c[r];
      if (mask) {
        int bb = row / QQ, qq = row % QQ;
        if (mask[bb * QQ + qq] != 0) val = 0.f;
      }
      Cout[(size_t)row * CC + col] = val;
    }
  }
}

// ---------------- Max over cameras + residual ReLU fuse ----------------
__global__ void fusemax_kernel(const float* __restrict__ v, const float* __restrict__ qp,
                               float* __restrict__ fused) {
  int tid = blockIdx.x * blockDim.x + threadIdx.x;
  if (tid >= BQ * CC) return;
  int c = tid % CC;
  int bq = tid / CC;
  int b = bq / QQ, q = bq % QQ;
  float mx = -INFINITY;
  #pragma unroll
  for (int n = 0; n < NCAM; n++)
    mx = fmaxf(mx, v[(size_t)((b * NCAM + n) * QQ + q) * CC + c]);
  float qv = qp[(size_t)bq * CC + c];
  float f = qv + mx;
  f = f > 0.f ? f : 0.f;
  fused[(size_t)bq * CC + c] = f + qv;
}

extern "C" void kernel_launch(void* const* d_in, const int* in_sizes, int n_in,
                              void* d_out, int out_size, void* d_ws, size_t ws_size,
                              hipStream_t stream) {
  const float* query = (const float*)d_in[0];
  const float* rp    = (const float*)d_in[1];
  const int*   mask  = (const int*)d_in[2];
  const float* intr  = (const float*)d_in[3];
  const float* extr  = (const float*)d_in[4];
  const float* f0    = (const float*)d_in[5];
  const float* f1    = (const float*)d_in[6];
  const float* f2    = (const float*)d_in[7];
  const float* f3    = (const float*)d_in[8];
  const float* Wq    = (const float*)d_in[9];
  const float* bq    = (const float*)d_in[10];
  const float* Wv    = (const float*)d_in[11];
  const float* bv    = (const float*)d_in[12];
  const float* Wo    = (const float*)d_in[13];
  const float* bo    = (const float*)d_in[14];
  float* out = (float*)d_out;

  // Workspace layout (floats): proj | qproj | ms | v | fused  (~26 MB total)
  float* ws    = (float*)d_ws;
  float4* proj = (float4*)ws;                       // BNQ * 4
  float* qproj = ws + (size_t)BNQ * 4;              // BQ * CC
  float* msb   = qproj + (size_t)BQ * CC;           // BNQ * CC
  float* vbuf  = msb + (size_t)BNQ * CC;            // BNQ * CC
  float* fused = vbuf + (size_t)BNQ * CC;           // BQ * CC

  proj_kernel<<<(BNQ + 255) / 256, 256, 0, stream>>>(rp, mask, intr, extr, proj);
  gemm_wmma<<<dim3((BQ + 15) / 16, 4), 128, 0, stream>>>(query, Wq, bq, qproj, BQ, nullptr);
  sample_kernel<<<BNQ, CC, 0, stream>>>(proj, f0, f1, f2, f3, msb);
  gemm_wmma<<<dim3(BNQ / 16, 4), 128, 0, stream>>>(msb, Wv, bv, vbuf, BNQ, nullptr);
  fusemax_kernel<<<(BQ * CC + 255) / 256, 256, 0, stream>>>(vbuf, qproj, fused);
  gemm_wmma<<<dim3((BQ + 15) / 16, 4), 128, 0, stream>>>(fused, Wo, bo, out, BQ, mask);
}